// Attention_69758858822101
// MI455X (gfx1250) — compile-verified
//
#include <hip/hip_runtime.h>
#include <math.h>

// ---------------------------------------------------------------------------
// Attention-LSTM decoder for MI455X (gfx1250), wave32, f32 WMMA path with
// async global->LDS double-buffered staging.
// Shapes hardcoded from setup_inputs(): B=256 T=64 D=512 H=512 E=256 C=6624
// S=26 steps=26.
// ---------------------------------------------------------------------------

typedef __attribute__((ext_vector_type(2))) float v2f;
typedef __attribute__((ext_vector_type(8))) float v8f;

#define WAVE 32

// LDS buffer geometry (floats): A tile 128x16 @ stride 20, W tile 64x16 @ 20
#define A_STRIDE   20
#define A_FLOATS   (128 * A_STRIDE)          // 2560
#define W_FLOATS   (64 * A_STRIDE)           // 1280
#define BUF_FLOATS (A_FLOATS + W_FLOATS)     // 3840 floats = 15360 B per buffer

__device__ __forceinline__ void async_load_b128(unsigned lds_off, const void* gaddr)
{
    // CDNA5 async copy: 16B global -> LDS, tracked by ASYNCcnt, no VGPR data.
    asm volatile("global_load_async_to_lds_b128 %0, %1, off"
                 :: "v"(lds_off), "v"(gaddr)
                 : "memory");
}

__device__ __forceinline__ void wait_async_all()
{
    asm volatile("s_wait_asynccnt 0x0" ::: "memory");
}

// ---------------------------------------------------------------------------
// GEMM: Out[M,N] = A[M,K] * W[N,K]^T (+bias[n]) (+Out if betaFlag)
// Block = 256 threads (8 wave32), 128x64 output tile, each wave owns 32x32.
// K staged 16-wide through double-buffered LDS via async b128 loads.
// Requirements (met by all call sites): M % 128 == 0, K % 16 == 0; N partial
// tiles handled by clamped loads (garbage never crosses columns) + store guard.
// ---------------------------------------------------------------------------
__global__ void __launch_bounds__(256)
gemm_tn_wmma_f32(const float* __restrict__ A, const float* __restrict__ W,
                 const float* __restrict__ bias, float* __restrict__ Out,
                 int M, int N, int K, int betaFlag)
{
    __shared__ float lds[2 * BUF_FLOATS];
    const unsigned ldsBase = (unsigned)(uintptr_t)(void*)lds;

    const int tid  = threadIdx.x;
    const int m0   = blockIdx.y * 128;
    const int n0   = blockIdx.x * 64;

    const int wid  = tid >> 5;          // 0..7
    const int lane = tid & (WAVE - 1);  // 0..31
    const int wm   = wid & 3;           // wave row tile: rows wm*32 .. +31
    const int wn   = wid >> 2;          // wave col tile: cols wn*32 .. +31
    const int col  = lane & 15;         // fragment M-row / N-col index
    const int khi  = lane >> 4;         // lane-half selector

    // cooperative-load coordinates (clamped: OOB rows load garbage that can
    // never reach a stored output element)
    const int lrowA0 = tid >> 2;                 // 0..63
    const int lrowA1 = lrowA0 + 64;              // 64..127
    const int lcol4  = (tid & 3) * 4;            // 0,4,8,12
    const int ga0    = min(m0 + lrowA0, M - 1);
    const int ga1    = min(m0 + lrowA1, M - 1);
    const int gw     = min(n0 + lrowA0, N - 1);

    const unsigned dstA0 = ldsBase + (unsigned)(lrowA0 * A_STRIDE + lcol4) * 4u;
    const unsigned dstA1 = ldsBase + (unsigned)(lrowA1 * A_STRIDE + lcol4) * 4u;
    const unsigned dstW  = ldsBase + (unsigned)(A_FLOATS + lrowA0 * A_STRIDE + lcol4) * 4u;
    const unsigned bufB  = (unsigned)BUF_FLOATS * 4u;

    v8f acc00 = {}, acc01 = {}, acc10 = {}, acc11 = {};

    // ---- prologue: fill buffer 0 ----
    async_load_b128(dstA0, A + (size_t)ga0 * K + lcol4);
    async_load_b128(dstA1, A + (size_t)ga1 * K + lcol4);
    async_load_b128(dstW,  W + (size_t)gw  * K + lcol4);
    wait_async_all();
    __syncthreads();

    int parity = 0;
    for (int k0 = 0; k0 < K; k0 += 16) {
        // ---- issue async loads for next slab into the other buffer ----
        if (k0 + 16 < K) {
            const unsigned off = (unsigned)(parity ^ 1) * bufB;
            const int kn = k0 + 16;
            async_load_b128(dstA0 + off, A + (size_t)ga0 * K + kn + lcol4);
            async_load_b128(dstA1 + off, A + (size_t)ga1 * K + kn + lcol4);
            async_load_b128(dstW  + off, W + (size_t)gw  * K + kn + lcol4);
        }

        // ---- compute on current buffer: 4 K-substeps x 4 WMMA ----
        const float* As = lds + (size_t)parity * BUF_FLOATS;
        const float* Ws = As + A_FLOATS;
        #pragma unroll
        for (int kk = 0; kk < 16; kk += 4) {
            const int kb = kk + 2 * khi;
            v2f a0, a1, b0, b1;
            // A 16x4 fragment layout: a[v] = A[row][v + 2*khi]
            a0.x = As[(wm * 32 + col) * A_STRIDE + kb + 0];
            a0.y = As[(wm * 32 + col) * A_STRIDE + kb + 1];
            a1.x = As[(wm * 32 + 16 + col) * A_STRIDE + kb + 0];
            a1.y = As[(wm * 32 + 16 + col) * A_STRIDE + kb + 1];
            // B 4x16 fragment layout: b[v] = B[v + 2*khi][col] = W[n][kb + v]
            b0.x = Ws[(wn * 32 + col) * A_STRIDE + kb + 0];
            b0.y = Ws[(wn * 32 + col) * A_STRIDE + kb + 1];
            b1.x = Ws[(wn * 32 + 16 + col) * A_STRIDE + kb + 0];
            b1.y = Ws[(wn * 32 + 16 + col) * A_STRIDE + kb + 1];
            acc00 = __builtin_amdgcn_wmma_f32_16x16x4_f32(
                        false, a0, false, b0, (short)0, acc00, false, false);
            acc01 = __builtin_amdgcn_wmma_f32_16x16x4_f32(
                        false, a0, false, b1, (short)0, acc01, false, false);
            acc10 = __builtin_amdgcn_wmma_f32_16x16x4_f32(
                        false, a1, false, b0, (short)0, acc10, false, false);
            acc11 = __builtin_amdgcn_wmma_f32_16x16x4_f32(
                        false, a1, false, b1, (short)0, acc11, false, false);
        }

        // our async loads for the next buffer must be visible before anyone
        // reads it; barrier also protects the buffer we just read
        wait_async_all();
        __syncthreads();
        parity ^= 1;
    }

    // ---- store: C/D layout -> M = r + 8*khi, N = col ----
    const int orA0 = m0 + wm * 32 + 8 * khi;   // rows of acc0x (+r)
    const int orA1 = orA0 + 16;                // rows of acc1x (+r)
    const int ocB0 = n0 + wn * 32 + col;       // cols of accx0
    const int ocB1 = ocB0 + 16;                // cols of accx1

    const float biasB0 = (bias && ocB0 < N) ? bias[ocB0] : 0.f;
    const float biasB1 = (bias && ocB1 < N) ? bias[ocB1] : 0.f;

    #pragma unroll
    for (int r = 0; r < 8; ++r) {
        if (ocB0 < N) {
            float* p0 = Out + (size_t)(orA0 + r) * N + ocB0;
            float* p1 = Out + (size_t)(orA1 + r) * N + ocB0;
            float v0 = acc00[r] + biasB0;
            float v1 = acc10[r] + biasB0;
            if (betaFlag) { v0 += *p0; v1 += *p1; }
            *p0 = v0; *p1 = v1;
        }
        if (ocB1 < N) {
            float* p0 = Out + (size_t)(orA0 + r) * N + ocB1;
            float* p1 = Out + (size_t)(orA1 + r) * N + ocB1;
            float v0 = acc01[r] + biasB1;
            float v1 = acc11[r] + biasB1;
            if (betaFlag) { v0 += *p0; v1 += *p1; }
            *p0 = v0; *p1 = v1;
        }
    }
}

// ---------------------------------------------------------------------------
// Fused attention step: one block (256 threads = 8 wave32) per batch element.
//   e[t]     = sum_d tanh(Hproj[b,t,d] + hp[b,d]) * w_score[d]
//   alpha    = softmax_t(e)
//   x[b,0:512]   = sum_t alpha[t] * batch_H[b,t,:]
//   x[b,512:768] = emb[text[b,step]]
// ---------------------------------------------------------------------------
__global__ void __launch_bounds__(256)
attn_step_kernel(const float* __restrict__ Hproj, const float* __restrict__ hp,
                 const float* __restrict__ w_score, const float* __restrict__ batch_H,
                 const float* __restrict__ emb, const int* __restrict__ text,
                 int step, float* __restrict__ x)
{
    __shared__ float hpS[512];
    __shared__ float wsS[512];
    __shared__ float eS[64];

    const int b    = blockIdx.x;
    const int tid  = threadIdx.x;
    const int wid  = tid >> 5;
    const int lane = tid & (WAVE - 1);

    for (int i = tid; i < 512; i += 256) {
        hpS[i] = hp[(size_t)b * 512 + i];
        wsS[i] = w_score[i];
    }
    __syncthreads();

    // scores: each wave handles 8 time steps
    const float* Hb = Hproj + (size_t)b * 64 * 512;
    #pragma unroll
    for (int tt = 0; tt < 8; ++tt) {
        const int t = wid * 8 + tt;
        const float* Hr = Hb + (size_t)t * 512;
        float p = 0.f;
        for (int d = lane; d < 512; d += WAVE)
            p += tanhf(Hr[d] + hpS[d]) * wsS[d];
        #pragma unroll
        for (int off = 16; off > 0; off >>= 1)
            p += __shfl_down(p, off, WAVE);
        if (lane == 0) eS[t] = p;
    }
    __syncthreads();

    // softmax over 64 values, first wave only
    if (tid < WAVE) {
        float e0 = eS[tid], e1 = eS[tid + 32];
        float m = fmaxf(e0, e1);
        #pragma unroll
        for (int off = 16; off > 0; off >>= 1)
            m = fmaxf(m, __shfl_down(m, off, WAVE));
        m = __shfl(m, 0, WAVE);
        float x0 = __expf(e0 - m), x1 = __expf(e1 - m);
        float s = x0 + x1;
        #pragma unroll
        for (int off = 16; off > 0; off >>= 1)
            s += __shfl_down(s, off, WAVE);
        s = __shfl(s, 0, WAVE);
        const float inv = 1.f / s;
        eS[tid]      = x0 * inv;
        eS[tid + 32] = x1 * inv;
    }
    __syncthreads();

    // context accumulation (coalesced over d for each t)
    const float* Bb = batch_H + (size_t)b * 64 * 512;
    for (int d = tid; d < 512; d += 256) {
        float s = 0.f;
        #pragma unroll 8
        for (int t = 0; t < 64; ++t)
            s = fmaf(eS[t], Bb[(size_t)t * 512 + d], s);
        x[(size_t)b * 768 + d] = s;
    }

    // character embedding into x[:,512:768]
    const int ch = text[b * 26 + step];
    for (int j = tid; j < 256; j += 256)
        x[(size_t)b * 768 + 512 + j] = emb[(size_t)ch * 256 + j];
}

// ---------------------------------------------------------------------------
// Pointwise LSTM cell update; records h_new into hs[b, step, :]
// gates layout: [i | f | g | o], each 512 wide.
// ---------------------------------------------------------------------------
__global__ void __launch_bounds__(256)
lstm_step_kernel(const float* __restrict__ gates, float* __restrict__ h,
                 float* __restrict__ c, float* __restrict__ hs, int step)
{
    const int idx = blockIdx.x * 256 + threadIdx.x;
    if (idx >= 256 * 512) return;
    const int b = idx >> 9;
    const int j = idx & 511;
    const float* g = gates + (size_t)b * 2048;
    const float gi = g[j];
    const float gf = g[512 + j];
    const float gg = g[1024 + j];
    const float go = g[1536 + j];
    const float si = 1.f / (1.f + __expf(-gi));
    const float sf = 1.f / (1.f + __expf(-gf));
    const float so = 1.f / (1.f + __expf(-go));
    const float cn = sf * c[idx] + si * tanhf(gg);
    const float hn = so * tanhf(cn);
    c[idx] = cn;
    h[idx] = hn;
    hs[((size_t)b * 26 + step) * 512 + j] = hn;
}

__global__ void __launch_bounds__(256)
zero_f32_kernel(float* __restrict__ p, int n)
{
    const int i = blockIdx.x * 256 + threadIdx.x;
    if (i < n) p[i] = 0.f;
}

// ---------------------------------------------------------------------------
extern "C" void kernel_launch(void* const* d_in, const int* in_sizes, int n_in,
                              void* d_out, int out_size, void* d_ws, size_t ws_size,
                              hipStream_t stream)
{
    (void)in_sizes; (void)n_in; (void)out_size; (void)ws_size;

    const float* batch_H = (const float*)d_in[0];   // (256,64,512)
    const int*   text    = (const int*)  d_in[1];   // (256,26)
    const float* W_i2h   = (const float*)d_in[2];   // (512,512)
    const float* W_h2h   = (const float*)d_in[3];   // (512,512)
    const float* b_h2h   = (const float*)d_in[4];   // (512)
    const float* w_score = (const float*)d_in[5];   // (512)
    const float* W_ih    = (const float*)d_in[6];   // (2048,768)
    const float* W_hh    = (const float*)d_in[7];   // (2048,512)
    const float* b_ih    = (const float*)d_in[8];   // (2048)
    const float* b_hh    = (const float*)d_in[9];   // (2048)
    const float* emb     = (const float*)d_in[10];  // (6624,256)
    const float* W_gen   = (const float*)d_in[11];  // (6624,512)
    const float* b_gen   = (const float*)d_in[12];  // (6624)
    float*       out     = (float*)d_out;           // (256,26,6624)

    // scratch partition (floats)
    float* ws    = (float*)d_ws;
    float* Hproj = ws;                          // 16384*512
    float* h     = Hproj + (size_t)16384 * 512; // 256*512
    float* c     = h     + (size_t)256 * 512;   // 256*512
    float* hp    = c     + (size_t)256 * 512;   // 256*512
    float* x     = hp    + (size_t)256 * 512;   // 256*768
    float* gates = x     + (size_t)256 * 768;   // 256*2048
    float* hs    = gates + (size_t)256 * 2048;  // 256*26*512

    const dim3 blk(256);

    // Hproj = batch_H @ W_i2h^T   : M=16384 N=512 K=512
    gemm_tn_wmma_f32<<<dim3(512 / 64, 16384 / 128), blk, 0, stream>>>(
        batch_H, W_i2h, nullptr, Hproj, 16384, 512, 512, 0);

    // h = c = 0  (contiguous)
    zero_f32_kernel<<<(2 * 256 * 512 + 255) / 256, blk, 0, stream>>>(h, 2 * 256 * 512);

    for (int s = 0; s < 26; ++s) {
        // hp = h @ W_h2h^T + b_h2h : M=256 N=512 K=512
        gemm_tn_wmma_f32<<<dim3(512 / 64, 256 / 128), blk, 0, stream>>>(
            h, W_h2h, b_h2h, hp, 256, 512, 512, 0);

        // attention + context + embedding -> x (256,768)
        attn_step_kernel<<<256, blk, 0, stream>>>(
            Hproj, hp, w_score, batch_H, emb, text, s, x);

        // gates = x @ W_ih^T + b_ih : M=256 N=2048 K=768
        gemm_tn_wmma_f32<<<dim3(2048 / 64, 256 / 128), blk, 0, stream>>>(
            x, W_ih, b_ih, gates, 256, 2048, 768, 0);

        // gates += h @ W_hh^T + b_hh : M=256 N=2048 K=512  (beta=1)
        gemm_tn_wmma_f32<<<dim3(2048 / 64, 256 / 128), blk, 0, stream>>>(
            h, W_hh, b_hh, gates, 256, 2048, 512, 1);

        // pointwise LSTM update; record h into hs[:, s, :]
        lstm_step_kernel<<<(256 * 512 + 255) / 256, blk, 0, stream>>>(
            gates, h, c, hs, s);
    }

    // probs = hs @ W_gen^T + b_gen : M=6656 N=6624 K=512
    gemm_tn_wmma_f32<<<dim3((6624 + 63) / 64, 6656 / 128), blk, 0, stream>>>(
        hs, W_gen, b_gen, out, 6656, 6624, 512, 0);
}